// ConvAttentionCoefficients_40046275067968
// MI455X (gfx1250) — compile-verified
//
#include <hip/hip_runtime.h>
#include <hip/hip_bf16.h>

// -------------------------------------------------------------------------
// out[p] = (1/sqrt(F)) * sum_f  (x@Wq)[idx_i[p],f] * w_ij[p,f] * (x@Wk)[idx_j[p],f]
// F = 128, f32 everywhere.  Bandwidth-bound: ~390 MB HBM traffic (~17us floor
// at 23.3 TB/s).  Phase 1: f32 WMMA projections into d_ws. Phase 2: streaming
// coalesced per-edge reduce.
// -------------------------------------------------------------------------

typedef __attribute__((ext_vector_type(2))) float v2f;
typedef __attribute__((ext_vector_type(8))) float v8f;

#define F_DIM 128
#define ROW_TILES 4           // 16-row tiles per wave -> 64 rows per block

// ---------------------------------------------------------------------------
// Kernel 1: q = x @ Wq, k = x @ Wk  via V_WMMA_F32_16X16X4_F32.
// Block = 256 threads = 8 wave32s; block covers a 64-row node slab.
// Wave w owns output columns [16w, 16w+16) and keeps 4 row-tiles x {Q,K}
// accumulators live, so each B fragment (Wq/Wk) load feeds 4 WMMAs.
//
// ISA 7.12.2 f32 fragment layouts (wave32):
//  A (16x4):  lanes 0-15: M=lane,   v0=K+0, v1=K+1
//             lanes16-31: M=lane-16, v0=K+2, v1=K+3
//  B (4x16):  lanes 0-15: N=lane,   v0=row K+0, v1=row K+1
//             lanes16-31: N=lane-16, v0=row K+2, v1=row K+3
//  C (16x16): VGPR r: lanes 0-15 -> (M=r, N=lane), lanes 16-31 -> (M=r+8,...)
// ---------------------------------------------------------------------------
__global__ __launch_bounds__(256) void qk_project_wmma(
    const float* __restrict__ x, const float* __restrict__ Wq,
    const float* __restrict__ Wk, float* __restrict__ q, float* __restrict__ k,
    int n_nodes) {
  const int lane = threadIdx.x & 31;
  const int wave = threadIdx.x >> 5;               // 0..7 -> column tile
  const int m0 = blockIdx.x * (16 * ROW_TILES);    // node row slab base
  if (m0 >= n_nodes) return;                       // uniform per block

  const int row   = lane & 15;                     // M (for A) / N (for B,C)
  const int khalf = (lane >> 4) * 2;               // 0 for lanes 0-15, 2 else
  const int n0    = wave * 16;

  // Per-tile A row pointers, clamped so EXEC stays all-ones in the WMMA loop
  // (ragged tails duplicate the last row; stores are guarded in slow path).
  const float* __restrict__ xrow[ROW_TILES];
#pragma unroll
  for (int t = 0; t < ROW_TILES; ++t) {
    int m = m0 + t * 16 + row;
    if (m >= n_nodes) m = n_nodes - 1;
    xrow[t] = x + (size_t)m * F_DIM;
  }

  v8f cq[ROW_TILES], ck[ROW_TILES];
#pragma unroll
  for (int t = 0; t < ROW_TILES; ++t) { cq[t] = (v8f){}; ck[t] = (v8f){}; }

  for (int kk = 0; kk < F_DIM; kk += 4) {
    const int ka = kk + khalf;
    // B fragments (shared by all ROW_TILES WMMAs): row-major (K, N)
    const size_t b0 = (size_t)ka * F_DIM + n0 + row;
    v2f bq, bk;
    bq.x = Wq[b0];
    bq.y = Wq[b0 + F_DIM];
    bk.x = Wk[b0];
    bk.y = Wk[b0 + F_DIM];
#pragma unroll
    for (int t = 0; t < ROW_TILES; ++t) {
      v2f a;
      a.x = xrow[t][ka];
      a.y = xrow[t][ka + 1];
      // (neg_a, A, neg_b, B, c_mod, C, reuse_a, reuse_b)
      cq[t] = __builtin_amdgcn_wmma_f32_16x16x4_f32(false, a, false, bq,
                                                    (short)0, cq[t], false,
                                                    false);
      ck[t] = __builtin_amdgcn_wmma_f32_16x16x4_f32(false, a, false, bk,
                                                    (short)0, ck[t], false,
                                                    false);
    }
  }

  // Store C tiles. Fast path: full slab (always taken when n_nodes % 64 == 0,
  // e.g. 40000 = 625*64) -> no per-row exec juggling.
  const int col = n0 + row;
  if (m0 + 16 * ROW_TILES <= n_nodes) {
#pragma unroll
    for (int t = 0; t < ROW_TILES; ++t) {
      const int mBase = m0 + t * 16 + (lane >> 4) * 8;
#pragma unroll
      for (int r = 0; r < 8; ++r) {
        q[(size_t)(mBase + r) * F_DIM + col] = cq[t][r];
        k[(size_t)(mBase + r) * F_DIM + col] = ck[t][r];
      }
    }
  } else {
#pragma unroll
    for (int t = 0; t < ROW_TILES; ++t) {
      const int mBase = m0 + t * 16 + (lane >> 4) * 8;
#pragma unroll
      for (int r = 0; r < 8; ++r) {
        const int mr = mBase + r;
        if (mr < n_nodes) {
          q[(size_t)mr * F_DIM + col] = cq[t][r];
          k[(size_t)mr * F_DIM + col] = ck[t][r];
        }
      }
    }
  }
}

// ---------------------------------------------------------------------------
// Kernel 2: per-edge triple-product reduce. One wave32 per edge; lane L loads
// float4 at element offset 4L -> each wave reads a fully-coalesced 512B line
// of w_ij (HBM stream) and of q[i]/k[j] (L2-resident: q+k = 41MB < 192MB L2).
// ---------------------------------------------------------------------------
__global__ __launch_bounds__(256) void edge_reduce(
    const float* __restrict__ q, const float* __restrict__ k,
    const float* __restrict__ w, const int* __restrict__ idx_i,
    const int* __restrict__ idx_j, float* __restrict__ out, int n_pairs) {
  const int lane = threadIdx.x & 31;
  const int edge = (int)((blockIdx.x * (unsigned)blockDim.x + threadIdx.x) >> 5);
  if (edge >= n_pairs) return;

  const int i = idx_i[edge];
  const int j = idx_j[edge];

  // Speculative prefetch of the w_ij stream a few edges ahead
  // (emits global_prefetch_b8; translation failures are silently dropped).
  if (edge + 8 < n_pairs)
    __builtin_prefetch(w + (size_t)(edge + 8) * F_DIM + lane * 4, 0, 0);

  const float4 wv = ((const float4*)(w + (size_t)edge * F_DIM))[lane];
  const float4 qv = ((const float4*)(q + (size_t)i * F_DIM))[lane];
  const float4 kv = ((const float4*)(k + (size_t)j * F_DIM))[lane];

  float s = qv.x * wv.x * kv.x + qv.y * wv.y * kv.y + qv.z * wv.z * kv.z +
            qv.w * wv.w * kv.w;

  // wave32 tree reduce
#pragma unroll
  for (int off = 16; off > 0; off >>= 1) s += __shfl_down(s, off, 32);

  if (lane == 0) out[edge] = s * 0.088388347648318447f;  // 1/sqrt(128)
}

// ---------------------------------------------------------------------------
extern "C" void kernel_launch(void* const* d_in, const int* in_sizes, int n_in,
                              void* d_out, int out_size, void* d_ws,
                              size_t ws_size, hipStream_t stream) {
  const float* x     = (const float*)d_in[0];
  const float* w_ij  = (const float*)d_in[1];
  const int*   idx_i = (const int*)d_in[2];
  const int*   idx_j = (const int*)d_in[3];
  const float* Wq    = (const float*)d_in[4];
  const float* Wk    = (const float*)d_in[5];
  float* out = (float*)d_out;

  const int n_nodes = in_sizes[0] / F_DIM;  // 40000
  const int n_pairs = in_sizes[2];          // 640000

  float* q = (float*)d_ws;
  float* k = q + (size_t)n_nodes * F_DIM;   // needs 2*n_nodes*F*4 = 41 MB

  // Phase 1: WMMA projections (64 rows per 256-thread block; 625 blocks)
  const int slabs = (n_nodes + 16 * ROW_TILES - 1) / (16 * ROW_TILES);
  qk_project_wmma<<<slabs, 256, 0, stream>>>(x, Wq, Wk, q, k, n_nodes);

  // Phase 2: streaming edge reduce (8 edges per 256-thread block)
  const int blocks = (n_pairs + 7) / 8;
  edge_reduce<<<blocks, 256, 0, stream>>>(q, k, w_ij, idx_i, idx_j, out,
                                          n_pairs);
}